// EGNN_46523085750930
// MI455X (gfx1250) — compile-verified
//
#include <hip/hip_runtime.h>
#include <math.h>

#define N_NODES 2500
#define E_EDGES 40000
#define HID     1280
#define MSG     24
#define EDGE_EMB 16
#define L_LAYERS 6
#define D_IN    2577          // 2*HID + EDGE_EMB + 1
#define NODE_IN (HID + MSG)   // 1304

typedef __attribute__((ext_vector_type(2))) float v2f;
typedef __attribute__((ext_vector_type(8))) float v8f;

#if defined(__AMDGCN__) && __has_builtin(__builtin_amdgcn_global_load_async_to_lds_b128)
#define USE_ASYNC_LDS 1
#else
#define USE_ASYNC_LDS 0
#endif

#if USE_ASYNC_LDS
typedef __attribute__((__vector_size__(4 * sizeof(int)))) int v4i_t;
__device__ __forceinline__ void async_copy_b128(const float* g, float* l)
{
    __builtin_amdgcn_global_load_async_to_lds_b128(
        (__attribute__((address_space(1))) v4i_t*)g,
        (__attribute__((address_space(3))) v4i_t*)l,
        /*offset=*/0, /*cpol=*/0);
}
template <int N>
__device__ __forceinline__ void wait_asynccnt()
{
#if __has_builtin(__builtin_amdgcn_s_wait_asynccnt)
    __builtin_amdgcn_s_wait_asynccnt(N);
#else
    asm volatile("s_wait_asynccnt %0" :: "n"(N) : "memory");
#endif
}
#endif

// ---------------------------------------------------------------------------
// Block-tiled fp32 WMMA GEMM (for K = multiple of 32):
//   C[M x Nw] = A[M x K] @ W[K x Nw] (+ bias)
// 256 threads = 8 waves, arranged 4(M) x 2(N) over a 64x32 C block tile.
// K-chunks of 32 staged in LDS, double-buffered via GLOBAL_LOAD_ASYNC_TO_LDS
// (ASYNCcnt + workgroup barrier), then V_WMMA_F32_16X16X4_F32 from LDS.
// NOTE: B tile may read up to 31 columns past Nw within the SAME parent
// tensor row block (always in-bounds for this problem's weight tensors);
// results for those columns are masked on store.
// ---------------------------------------------------------------------------
#define BM 64
#define BN 32
#define BK 32

__global__ __launch_bounds__(256)
void gemm_wmma_tiled(const float* __restrict__ A, const float* __restrict__ W,
                     const float* __restrict__ bias, float* __restrict__ C,
                     int M, int K, int Nw, int ldA, int ldW, int ldC)
{
    __shared__ __align__(16) float sA[2][BM * BK];
    __shared__ __align__(16) float sB[2][BK * BN];

    const int tid  = threadIdx.x;
    const int lane = tid & 31;
    const int w    = tid >> 5;       // wave 0..7
    const int wm   = w & 3;          // M sub-tile 0..3
    const int wn   = w >> 2;         // N sub-tile 0..1
    const int half = lane >> 4;
    const int l16  = lane & 15;
    const int bm = blockIdx.x, bn = blockIdx.y;

    // Cooperative staging indices: 4-float (16B) transfers.
    const int arow = tid >> 3;            // 0..31 (and +32 for 2nd xfer)
    const int acol = (tid & 7) * 4;
    const int brow = tid >> 3;            // 0..31
    const int bcol = (tid & 7) * 4;

    const long Abase = (long)bm * BM;
    long mA0 = Abase + arow;       if (mA0 >= M) mA0 = M - 1;   // clamp OOB rows
    long mA1 = Abase + arow + 32;  if (mA1 >= M) mA1 = M - 1;
    const float* __restrict__ gA0 = A + mA0 * ldA + acol;
    const float* __restrict__ gA1 = A + mA1 * ldA + acol;
    const float* __restrict__ gB  = W + (long)brow * ldW + (long)bn * BN + bcol;

    v8f acc = {0.f, 0.f, 0.f, 0.f, 0.f, 0.f, 0.f, 0.f};
    const int NC = K / BK;

    const float* __restrict__ pAr = &sA[0][(wm * 16 + l16) * BK];
    const float* __restrict__ pBr = &sB[0][wn * 16 + l16];
    const int bufStrideA = BM * BK;
    const int bufStrideB = BK * BN;

#if USE_ASYNC_LDS
    // ---- async double-buffered pipeline ----
    {   // prefetch chunk 0 into buffer 0
        async_copy_b128(gA0, &sA[0][arow * BK + acol]);
        async_copy_b128(gA1, &sA[0][(arow + 32) * BK + acol]);
        async_copy_b128(gB,  &sB[0][brow * BN + bcol]);
    }
    for (int c = 0; c < NC; ++c) {
        const int cur = c & 1;
        if (c + 1 < NC) {
            const int nxt = cur ^ 1;
            const int kb  = (c + 1) * BK;
            async_copy_b128(gA0 + kb, &sA[nxt][arow * BK + acol]);
            async_copy_b128(gA1 + kb, &sA[nxt][(arow + 32) * BK + acol]);
            async_copy_b128(gB + (long)kb * ldW, &sB[nxt][brow * BN + bcol]);
            wait_asynccnt<3>();      // chunk c complete (in-order), c+1 in flight
        } else {
            wait_asynccnt<0>();
        }
        __syncthreads();
        const float* pA = pAr + cur * bufStrideA;
        const float* pB = pBr + cur * bufStrideB;
        #pragma unroll
        for (int kk = 0; kk < BK; kk += 4) {
            const int k0 = kk + 2 * half;
            v2f a; a.x = pA[k0];            a.y = pA[k0 + 1];
            v2f b; b.x = pB[k0 * BN];       b.y = pB[(k0 + 1) * BN];
            acc = __builtin_amdgcn_wmma_f32_16x16x4_f32(
                false, a, false, b, (short)0, acc, false, false);
        }
        __syncthreads();             // buffer `cur` free for reuse at c+2
    }
#else
    // ---- synchronous single-buffer fallback ----
    for (int c = 0; c < NC; ++c) {
        const int kb = c * BK;
        __syncthreads();
        #pragma unroll
        for (int j = 0; j < 4; ++j) {
            sA[0][arow * BK + acol + j]        = gA0[kb + j];
            sA[0][(arow + 32) * BK + acol + j] = gA1[kb + j];
            sB[0][brow * BN + bcol + j]        = gB[(long)kb * ldW + j];
        }
        __syncthreads();
        #pragma unroll
        for (int kk = 0; kk < BK; kk += 4) {
            const int k0 = kk + 2 * half;
            v2f a; a.x = pAr[k0];           a.y = pAr[k0 + 1];
            v2f b; b.x = pBr[k0 * BN];      b.y = pBr[(k0 + 1) * BN];
            acc = __builtin_amdgcn_wmma_f32_16x16x4_f32(
                false, a, false, b, (short)0, acc, false, false);
        }
    }
#endif

    const int n_out = bn * BN + wn * 16 + l16;
    if (n_out < Nw) {
        const float bv = bias ? bias[n_out] : 0.0f;
        #pragma unroll
        for (int r = 0; r < 8; ++r) {
            const int m = bm * BM + wm * 16 + r + 8 * half;
            if (m < M) C[(long)m * ldC + n_out] = acc[r] + bv;
        }
    }
}

// ---------------------------------------------------------------------------
// Wave-per-tile fp32 WMMA GEMM for small K (used for K=16 edge projection).
// ---------------------------------------------------------------------------
__global__ __launch_bounds__(32)
void gemm_wmma_f32(const float* __restrict__ A, const float* __restrict__ W,
                   const float* __restrict__ bias, float* __restrict__ C,
                   int M, int K, int Nw, int ldA, int ldW, int ldC)
{
    const int lane = threadIdx.x & 31;
    const int half = lane >> 4;
    const int l16  = lane & 15;
    const int bm = blockIdx.x, bn = blockIdx.y;

    const int m_frag = bm * 16 + l16;
    const int n_frag = bn * 16 + l16;
    const int mA = (m_frag < M)  ? m_frag : (M - 1);
    const int nB = (n_frag < Nw) ? n_frag : (Nw - 1);

    const float* __restrict__ Arow = A + (long)mA * ldA;
    const float* __restrict__ Bcol = W + nB;

    v8f acc = {0.f, 0.f, 0.f, 0.f, 0.f, 0.f, 0.f, 0.f};

    for (int kb = 0; kb < K; kb += 4) {
        const int k0 = kb + 2 * half;
        v2f a; a.x = Arow[k0];                 a.y = Arow[k0 + 1];
        v2f b; b.x = Bcol[(long)k0 * ldW];     b.y = Bcol[(long)(k0 + 1) * ldW];
        acc = __builtin_amdgcn_wmma_f32_16x16x4_f32(
            false, a, false, b, (short)0, acc, false, false);
    }

    if (n_frag < Nw) {
        const float bv = bias ? bias[n_frag] : 0.0f;
        #pragma unroll
        for (int r = 0; r < 8; ++r) {
            const int m = bm * 16 + r + 8 * half;
            if (m < M) C[(long)m * ldC + n_frag] = acc[r] + bv;
        }
    }
}

// ---------------------------------------------------------------------------
__global__ __launch_bounds__(256)
void edge_attr_kernel(const float* __restrict__ pos0, const int* __restrict__ row,
                      const int* __restrict__ col, const float* __restrict__ freqs,
                      float* __restrict__ ea)
{
    const int e = blockIdx.x * blockDim.x + threadIdx.x;
    if (e >= E_EDGES) return;
    const int r = row[e], c = col[e];
    const float dx = pos0[r*3+0] - pos0[c*3+0];
    const float dy = pos0[r*3+1] - pos0[c*3+1];
    const float dz = pos0[r*3+2] - pos0[c*3+2];
    const float d  = sqrtf(dx*dx + dy*dy + dz*dz);
    #pragma unroll
    for (int k = 0; k < EDGE_EMB/2; ++k) {
        const float ang = d * freqs[k];
        ea[e*EDGE_EMB + 2*k]     = sinf(ang);
        ea[e*EDGE_EMB + 2*k + 1] = cosf(ang);
    }
}

__global__ __launch_bounds__(256)
void zero_kernel(float* __restrict__ p, int n)
{
    const int i = blockIdx.x * blockDim.x + threadIdx.x;
    if (i < n) p[i] = 0.0f;
}

__global__ __launch_bounds__(256)
void copyf_kernel(const float* __restrict__ src, float* __restrict__ dst, int n)
{
    const int i = blockIdx.x * blockDim.x + threadIdx.x;
    if (i < n) dst[i] = src[i];
}

__device__ __forceinline__ float silu_f(float x) { return x / (1.0f + expf(-x)); }

// ---------------------------------------------------------------------------
// Per-edge message MLP + scatter (aggregation index = col).
// ---------------------------------------------------------------------------
__global__ __launch_bounds__(256)
void edge_msg_kernel(const int* __restrict__ row, const int* __restrict__ col,
                     const float* __restrict__ pos,
                     const float* __restrict__ P, const float* __restrict__ Q,
                     const float* __restrict__ eC, const float* __restrict__ wr,
                     const float* __restrict__ W2, const float* __restrict__ b2,
                     const float* __restrict__ cW1, const float* __restrict__ cb1,
                     const float* __restrict__ cW2, int update_coords,
                     float* __restrict__ msg_sum, float* __restrict__ cnt,
                     float* __restrict__ coord_agg)
{
    __shared__ float sW2[MSG*MSG], scW1[MSG*MSG];
    __shared__ float sb2[MSG], scb1[MSG], scW2[MSG], swr[MSG];
    const int tid = threadIdx.x;
    for (int i = tid; i < MSG*MSG; i += blockDim.x) {
        sW2[i]  = W2[i];
        scW1[i] = update_coords ? cW1[i] : 0.0f;
    }
    if (tid < MSG) {
        sb2[tid]  = b2[tid];
        swr[tid]  = wr[tid];
        scb1[tid] = update_coords ? cb1[tid] : 0.0f;
        scW2[tid] = update_coords ? cW2[tid] : 0.0f;
    }
    __syncthreads();

    const int e = blockIdx.x * blockDim.x + tid;
    if (e >= E_EDGES) return;
    const int r = row[e], c = col[e];
    const float cd0 = pos[r*3+0] - pos[c*3+0];
    const float cd1 = pos[r*3+1] - pos[c*3+1];
    const float cd2 = pos[r*3+2] - pos[c*3+2];
    const float radial = cd0*cd0 + cd1*cd1 + cd2*cd2;

    float m1[MSG];
    #pragma unroll
    for (int j = 0; j < MSG; ++j) {
        const float v = P[c*MSG+j] + Q[r*MSG+j] + eC[(long)e*MSG+j] + radial * swr[j];
        m1[j] = silu_f(v);
    }
    float m[MSG];
    #pragma unroll
    for (int j = 0; j < MSG; ++j) {
        float v = sb2[j];
        #pragma unroll
        for (int k = 0; k < MSG; ++k) v += m1[k] * sW2[k*MSG+j];
        m[j] = silu_f(v);
    }
    #pragma unroll
    for (int j = 0; j < MSG; ++j) atomicAdd(&msg_sum[c*MSG+j], m[j]);
    atomicAdd(&cnt[c], 1.0f);

    if (update_coords) {
        float cw = 0.0f;
        #pragma unroll
        for (int j = 0; j < MSG; ++j) {
            float v = scb1[j];
            #pragma unroll
            for (int k = 0; k < MSG; ++k) v += m[k] * scW1[k*MSG+j];
            cw += silu_f(v) * scW2[j];
        }
        atomicAdd(&coord_agg[c*3+0], cd0 * cw);
        atomicAdd(&coord_agg[c*3+1], cd1 * cw);
        atomicAdd(&coord_agg[c*3+2], cd2 * cw);
    }
}

// ---------------------------------------------------------------------------
// Per-node update + LayerNorm. One block (256 threads) per node; 1280 = 256*5.
// ---------------------------------------------------------------------------
__global__ __launch_bounds__(256)
void node_finish_kernel(float* __restrict__ h, float* __restrict__ pos,
                        const float* __restrict__ R, const float* __restrict__ nW1b,
                        const float* __restrict__ nW2, const float* __restrict__ nb2,
                        const float* __restrict__ g, const float* __restrict__ b,
                        const float* __restrict__ msg_sum, const float* __restrict__ cnt,
                        const float* __restrict__ coord_agg, int update_coords)
{
    const int n = blockIdx.x;
    const int tid = threadIdx.x;
    __shared__ float magg[MSG], t[MSG];
    __shared__ float red[256], red2[256];

    if (tid < MSG) magg[tid] = msg_sum[n*MSG+tid] / fmaxf(cnt[n], 1.0f);
    __syncthreads();
    if (tid < MSG) {
        float v = R[n*MSG+tid];
        #pragma unroll
        for (int k = 0; k < MSG; ++k) v += magg[k] * nW1b[k*MSG+tid];
        t[tid] = silu_f(v);
    }
    __syncthreads();

    float xs[5];
    float s = 0.0f, s2 = 0.0f;
    #pragma unroll
    for (int i = 0; i < 5; ++i) {
        const int d = tid + i * 256;
        float hu = nb2[d];
        #pragma unroll
        for (int j = 0; j < MSG; ++j) hu += t[j] * nW2[(long)j*HID + d];
        const float x = h[(long)n*HID + d] + hu;
        xs[i] = x; s += x; s2 += x * x;
    }
    red[tid] = s; red2[tid] = s2;
    __syncthreads();
    for (int off = 128; off > 0; off >>= 1) {
        if (tid < off) { red[tid] += red[tid+off]; red2[tid] += red2[tid+off]; }
        __syncthreads();
    }
    const float mu  = red[0] / (float)HID;
    const float var = red2[0] / (float)HID - mu * mu;
    const float rs  = rsqrtf(var + 1e-5f);
    #pragma unroll
    for (int i = 0; i < 5; ++i) {
        const int d = tid + i * 256;
        h[(long)n*HID + d] = (xs[i] - mu) * rs * g[d] + b[d];
    }
    if (update_coords && tid < 3) pos[n*3+tid] += coord_agg[n*3+tid];
}

__global__ __launch_bounds__(256)
void copy_out_kernel(const float* __restrict__ h, const float* __restrict__ pos,
                     float* __restrict__ out)
{
    const int i = blockIdx.x * blockDim.x + threadIdx.x;
    const int nh = N_NODES * HID;
    if (i < nh) out[i] = h[i];
    else if (i < nh + N_NODES * 3) out[i] = pos[i - nh];
}

// ---------------------------------------------------------------------------
extern "C" void kernel_launch(void* const* d_in, const int* in_sizes, int n_in,
                              void* d_out, int out_size, void* d_ws, size_t ws_size,
                              hipStream_t stream)
{
    const float* h_in   = (const float*)d_in[0];
    const float* pos_in = (const float*)d_in[1];
    const int*   eidx   = (const int*)  d_in[2];
    const float* nembW  = (const float*)d_in[3];
    const float* nembB  = (const float*)d_in[4];
    const float* msgW1  = (const float*)d_in[5];
    const float* msgB1  = (const float*)d_in[6];
    const float* msgW2  = (const float*)d_in[7];
    const float* msgB2  = (const float*)d_in[8];
    const float* cW1    = (const float*)d_in[9];
    const float* cB1    = (const float*)d_in[10];
    const float* cW2    = (const float*)d_in[11];
    const float* nodeW1 = (const float*)d_in[12];
    const float* nodeB1 = (const float*)d_in[13];
    const float* nodeW2 = (const float*)d_in[14];
    const float* nodeB2 = (const float*)d_in[15];
    const float* lnG    = (const float*)d_in[16];
    const float* lnB    = (const float*)d_in[17];
    const float* freqs  = (const float*)d_in[18];

    const int* row = eidx;
    const int* col = eidx + E_EDGES;

    // Workspace layout (fp32), ~20.3 MB total — all L2-resident.
    float* ws   = (float*)d_ws;
    float* h    = ws;  ws += N_NODES * HID;
    float* pos  = ws;  ws += N_NODES * 4;
    float* ea   = ws;  ws += E_EDGES * EDGE_EMB;
    float* eC   = ws;  ws += E_EDGES * MSG;
    float* P    = ws;  ws += N_NODES * MSG;
    float* Q    = ws;  ws += N_NODES * MSG;
    float* R    = ws;  ws += N_NODES * MSG;
    float* msum = ws;  ws += N_NODES * MSG;
    float* cnt  = ws;  ws += N_NODES;
    float* cagg = ws;  ws += N_NODES * 4;

    const int MB = (N_NODES + BM - 1) / BM;   // 40 row blocks of 64
    const int ET = E_EDGES / 16;              // 2500 edge row-tiles
    const dim3 blk32(32), blk256(256);

    copyf_kernel<<<(N_NODES*3 + 255)/256, blk256, 0, stream>>>(pos_in, pos, N_NODES*3);
    edge_attr_kernel<<<(E_EDGES + 255)/256, blk256, 0, stream>>>(pos_in, row, col, freqs, ea);

    // h = h_in @ node_emb_W + b   (2500x1280 @ 1280x1280)
    gemm_wmma_tiled<<<dim3(MB, HID/BN), blk256, 0, stream>>>(
        h_in, nembW, nembB, h, N_NODES, HID, HID, HID, HID, HID);

    for (int i = 0; i < L_LAYERS; ++i) {
        const int upd = (i < L_LAYERS - 1) ? 1 : 0;
        const float* W1   = msgW1 + (long)i * D_IN * MSG;
        const float* W1b  = W1 + (long)HID * MSG;        // rows for h[row]
        const float* W1e  = W1 + (long)2 * HID * MSG;    // rows for edge_attr
        const float* W1r  = W1 + (long)(D_IN - 1) * MSG; // radial row
        const float* nW1  = nodeW1 + (long)i * NODE_IN * MSG;
        const float* nW1b = nW1 + (long)HID * MSG;
        const int ci = upd ? i : (L_LAYERS - 2);         // valid (unused) ptrs last layer

        zero_kernel<<<(N_NODES*MSG + 255)/256, blk256, 0, stream>>>(msum, N_NODES*MSG);
        zero_kernel<<<(N_NODES   + 255)/256, blk256, 0, stream>>>(cnt,  N_NODES);
        zero_kernel<<<(N_NODES*3 + 255)/256, blk256, 0, stream>>>(cagg, N_NODES*3);

        // Node-side projections of the edge-MLP input (concat distributivity):
        gemm_wmma_tiled<<<dim3(MB, 1), blk256, 0, stream>>>(   // P = h@W1[:1280] + b1
            h, W1, msgB1 + (long)i*MSG, P, N_NODES, HID, MSG, HID, MSG, MSG);
        gemm_wmma_tiled<<<dim3(MB, 1), blk256, 0, stream>>>(   // Q = h@W1[1280:2560]
            h, W1b, nullptr, Q, N_NODES, HID, MSG, HID, MSG, MSG);
        gemm_wmma_f32<<<dim3(ET, 2), blk32, 0, stream>>>(      // eC = ea@W1[2560:2576]
            ea, W1e, nullptr, eC, E_EDGES, EDGE_EMB, MSG, EDGE_EMB, MSG, MSG);
        gemm_wmma_tiled<<<dim3(MB, 1), blk256, 0, stream>>>(   // R = h@nodeW1[:1280]+nb1
            h, nW1, nodeB1 + (long)i*MSG, R, N_NODES, HID, MSG, HID, MSG, MSG);

        edge_msg_kernel<<<(E_EDGES + 255)/256, blk256, 0, stream>>>(
            row, col, pos, P, Q, eC, W1r,
            msgW2 + (long)i*MSG*MSG, msgB2 + (long)i*MSG,
            cW1 + (long)ci*MSG*MSG, cB1 + (long)ci*MSG, cW2 + (long)ci*MSG,
            upd, msum, cnt, cagg);

        node_finish_kernel<<<N_NODES, blk256, 0, stream>>>(
            h, pos, R, nW1b,
            nodeW2 + (long)i*MSG*HID, nodeB2 + (long)i*HID,
            lnG + (long)i*HID, lnB + (long)i*HID,
            msum, cnt, cagg, upd);
    }

    copy_out_kernel<<<(N_NODES*HID + N_NODES*3 + 255)/256, blk256, 0, stream>>>(h, pos, (float*)d_out);
}